// GreaseLMEncoder_89781996356179
// MI455X (gfx1250) — compile-verified
//
#include <hip/hip_runtime.h>
#include <hip/hip_bf16.h>
#include <math.h>

#define EMB 128
#define ETY 38
#define NTY 4
#define NCOMBO 624   /* 39 * 4 * 4 */
#define HEADS 4
#define KCHUNK 64
#define BT_STRIDE 68 /* 68*c + k -> banks (4c+k)%64: conflict-free b64 reads, 8B aligned */

typedef float v2f __attribute__((ext_vector_type(2)));
typedef float v8f __attribute__((ext_vector_type(8)));

// ---------------------------------------------------------------------------
// helpers
// ---------------------------------------------------------------------------
__device__ __forceinline__ void atomic_max_f32(float* addr, float val) {
  // sign-split trick: monotone mapping of float order onto int/uint order
  if (val >= 0.0f) atomicMax((int*)addr, __float_as_int(val));
  else             atomicMin((unsigned int*)addr, __float_as_uint(val));
}

__device__ __forceinline__ int edge_src(int e, const int* __restrict__ s0, int E) {
  return e < E ? s0[e] : e - E;
}
__device__ __forceinline__ int edge_dst(int e, const int* __restrict__ d0, int E) {
  return e < E ? d0[e] : e - E;
}

// ---------------------------------------------------------------------------
// 0: zero / -inf init of per-call accumulators
// ---------------------------------------------------------------------------
__global__ void init_kernel(float* __restrict__ cnt, float* __restrict__ smax,
                            float* __restrict__ denom, float* __restrict__ aggr,
                            float* __restrict__ ccount, float* __restrict__ osum,
                            float* __restrict__ osumsq, int N) {
  int idx = blockIdx.x * blockDim.x + threadIdx.x;
  if (idx < N * EMB)   aggr[idx] = 0.0f;
  if (idx < N * HEADS) { smax[idx] = -INFINITY; denom[idx] = 0.0f; }
  if (idx < N)         cnt[idx] = 0.0f;
  if (idx < NCOMBO)    ccount[idx] = 0.0f;
  if (idx < EMB)       { osum[idx] = 0.0f; osumsq[idx] = 0.0f; }
}

// ---------------------------------------------------------------------------
// 1: per-edge combo id, combo histogram, src-degree count
// ---------------------------------------------------------------------------
__global__ void combo_kernel(const int* __restrict__ etype, const int* __restrict__ ntype,
                             const int* __restrict__ src0, const int* __restrict__ dst0,
                             int* __restrict__ combo, float* __restrict__ ccount,
                             float* __restrict__ cnt, int E, int Et) {
  int e = blockIdx.x * blockDim.x + threadIdx.x;
  if (e >= Et) return;
  int et, s, t, sn;
  if (e < E) {
    et = etype[e];
    sn = src0[e];
    s = ntype[sn];
    t = ntype[dst0[e]];
  } else {
    int i = e - E;
    et = ETY;          // self-loop type = 38
    sn = i;
    s = ntype[i];
    t = s;
  }
  int c = et * 16 + s * 4 + t;
  combo[e] = c;
  atomicAdd(&ccount[c], 1.0f);
  atomicAdd(&cnt[sn], 1.0f);
}

// ---------------------------------------------------------------------------
// 2: h1 table = feats @ We1 + be1 per combo (one-hot => 3 row gathers)
// ---------------------------------------------------------------------------
__global__ void h1_table_kernel(const float* __restrict__ We1, const float* __restrict__ be1,
                                float* __restrict__ h1t) {
  int idx = blockIdx.x * blockDim.x + threadIdx.x;
  if (idx >= NCOMBO * EMB) return;
  int c = idx >> 7, col = idx & 127;
  int et = c >> 4, s = (c >> 2) & 3, t = c & 3;
  h1t[idx] = We1[et * EMB + col] + We1[(ETY + 1 + s) * EMB + col] +
             We1[(ETY + 1 + NTY + t) * EMB + col] + be1[col];
}

// ---------------------------------------------------------------------------
// 3: exact BN batch stats over Et rows via combo-count weighting (1 block)
// ---------------------------------------------------------------------------
__global__ void edge_bn_stats_kernel(const float* __restrict__ h1t,
                                     const float* __restrict__ ccount,
                                     float* __restrict__ emean, float* __restrict__ erstd,
                                     int Et) {
  int col = threadIdx.x;  // 128 threads
  float s = 0.0f, ss = 0.0f;
  for (int c = 0; c < NCOMBO; ++c) {
    float w = ccount[c];
    float v = h1t[c * EMB + col];
    s += w * v;
    ss += w * v * v;
  }
  float inv = 1.0f / (float)Et;
  float m = s * inv;
  float var = ss * inv - m * m;
  var = var > 0.0f ? var : 0.0f;
  emean[col] = m;
  erstd[col] = rsqrtf(var + 1e-5f);
}

__global__ void edge_bn_apply_kernel(const float* __restrict__ h1t,
                                     const float* __restrict__ emean,
                                     const float* __restrict__ erstd,
                                     const float* __restrict__ ge,
                                     const float* __restrict__ bbe,
                                     float* __restrict__ bnt) {
  int idx = blockIdx.x * blockDim.x + threadIdx.x;
  if (idx >= NCOMBO * EMB) return;
  int col = idx & 127;
  float v = (h1t[idx] - emean[col]) * erstd[col] * ge[col] + bbe[col];
  bnt[idx] = v > 0.0f ? v : 0.0f;
}

// ---------------------------------------------------------------------------
// WMMA f32 GEMM:  C[M,128] = (A[M,K] @ B[K,128] + bias) * scale
// A may be split into two [M,128] halves (A = cols 0..127, A2 = cols 128..255).
// One wave computes a 16-row x 128-col stripe; M must be a multiple of 16.
// B is staged per 64-row K-chunk into LDS, transposed + padded, so each lane
// fetches its (B[kk][c], B[kk+1][c]) fragment as one conflict-free ds_load_b64.
// Uses V_WMMA_F32_16X16X4_F32 (full f32 precision).
// ---------------------------------------------------------------------------
__global__ __launch_bounds__(256) void wmma_gemm_kernel(
    const float* __restrict__ A, const float* __restrict__ A2, int lda,
    const float* __restrict__ B, const float* __restrict__ bias, float scale,
    float* __restrict__ C, int M, int K) {
  __shared__ float Bt[EMB * BT_STRIDE];  // 34,816 B

  const int wave = threadIdx.x >> 5;
  const int lane = threadIdx.x & 31;
  const int tile = blockIdx.x * 8 + wave;
  const int ntiles = M >> 4;
  const bool active = tile < ntiles;     // wave-uniform: EXEC all-1s inside WMMA

  const int row_base = tile << 4;
  const int m_l = lane & 15;
  const int sel = lane >> 4;             // half-wave selector
  const int row = row_base + m_l;

  v8f acc[8] = {};

  for (int kbase = 0; kbase < K; kbase += KCHUNK) {
    // cooperative transpose-staging of B[kbase..kbase+63][0..127] into LDS
#pragma unroll
    for (int i = 0; i < 8; ++i) {
      int base = (threadIdx.x + i * 256) * 4;  // element index within 64x128 chunk
      int c = base & 127;                      // column (multiple of 4)
      int k = base >> 7;                       // k within chunk
      const float4 v = *(const float4*)(B + (size_t)(kbase + k) * EMB + c);
      Bt[(c + 0) * BT_STRIDE + k] = v.x;
      Bt[(c + 1) * BT_STRIDE + k] = v.y;
      Bt[(c + 2) * BT_STRIDE + k] = v.z;
      Bt[(c + 3) * BT_STRIDE + k] = v.w;
    }
    __syncthreads();

    if (active) {
      for (int k0 = 0; k0 < KCHUNK; k0 += 4) {
        int kk = kbase + k0 + 2 * sel;         // A/B: lane holds K = kk, kk+1
        const float* Ap = (A2 && kk >= 128) ? A2 : A;
        int kc = (A2 && kk >= 128) ? kk - 128 : kk;
        v2f a = *(const v2f*)(Ap + (size_t)row * lda + kc);
        int kl = k0 + 2 * sel;                 // k within chunk (even -> 8B aligned)
#pragma unroll
        for (int n = 0; n < 8; ++n) {
          v2f b = *(const v2f*)(Bt + (n * 16 + m_l) * BT_STRIDE + kl);
          acc[n] = __builtin_amdgcn_wmma_f32_16x16x4_f32(
              false, a, false, b, (short)0, acc[n], false, false);
        }
      }
    }
    __syncthreads();
  }

  if (active) {
#pragma unroll
    for (int n = 0; n < 8; ++n) {
      int col = n * 16 + m_l;
      float bv = bias ? bias[col] : 0.0f;
#pragma unroll
      for (int j = 0; j < 8; ++j) {      // D VGPR j holds rows j (lanes 0-15) / j+8
        int r = row_base + j + 8 * sel;
        C[(size_t)r * EMB + col] = (acc[n][j] + bv) * scale;
      }
    }
  }
}

// ---------------------------------------------------------------------------
// 4: attention scores (wave per edge) + running segment max over src
// ---------------------------------------------------------------------------
__global__ __launch_bounds__(256) void scores_kernel(
    const float* __restrict__ Qn, const float* __restrict__ Kn,
    const float* __restrict__ ekb, const int* __restrict__ combo,
    const int* __restrict__ src0, const int* __restrict__ dst0,
    float* __restrict__ scores, float* __restrict__ smax, int E, int Et) {
  int e = (blockIdx.x * blockDim.x + threadIdx.x) >> 5;
  int lane = threadIdx.x & 31;
  if (e >= Et) return;
  int s = edge_src(e, src0, E);
  int d = edge_dst(e, dst0, E);
  int c = combo[e];

  const float4 q = *(const float4*)(Qn + (size_t)s * EMB + 4 * lane);
  const float4 kn = *(const float4*)(Kn + (size_t)d * EMB + 4 * lane);
  const float4 ke = *(const float4*)(ekb + (size_t)c * EMB + 4 * lane);
  float p = q.x * (kn.x + ke.x) + q.y * (kn.y + ke.y) +
            q.z * (kn.z + ke.z) + q.w * (kn.w + ke.w);
  // reduce within each 8-lane head group (head = lane>>3, 32 dims/head)
  p += __shfl_xor(p, 1, 32);
  p += __shfl_xor(p, 2, 32);
  p += __shfl_xor(p, 4, 32);
  if ((lane & 7) == 0) {
    int h = lane >> 3;
    scores[(size_t)e * HEADS + h] = p;
    atomic_max_f32(&smax[s * HEADS + h], p);
  }
}

// ---------------------------------------------------------------------------
// 5: ex = exp(score - smax[src]); denom accumulation (in-place into scores)
// ---------------------------------------------------------------------------
__global__ void exp_kernel(float* __restrict__ scores, const float* __restrict__ smax,
                           float* __restrict__ denom, const int* __restrict__ src0,
                           int E, int Et) {
  int tid = blockIdx.x * blockDim.x + threadIdx.x;
  if (tid >= Et * HEADS) return;
  int e = tid >> 2, h = tid & 3;
  int s = edge_src(e, src0, E);
  float ex = expf(scores[tid] - smax[s * HEADS + h]);
  scores[tid] = ex;
  atomicAdd(&denom[s * HEADS + h], ex);
}

// ---------------------------------------------------------------------------
// 6: alpha * message, scatter-add to aggr[dst] (wave per edge)
// ---------------------------------------------------------------------------
__global__ __launch_bounds__(256) void aggregate_kernel(
    const float* __restrict__ Mn, const float* __restrict__ emb,
    const int* __restrict__ combo, const float* __restrict__ ex,
    const float* __restrict__ denom, const float* __restrict__ cnt,
    const int* __restrict__ src0, const int* __restrict__ dst0,
    float* __restrict__ aggr, int E, int Et) {
  int e = (blockIdx.x * blockDim.x + threadIdx.x) >> 5;
  int lane = threadIdx.x & 31;
  if (e >= Et) return;
  int s = edge_src(e, src0, E);
  int d = edge_dst(e, dst0, E);
  int c = combo[e];
  int h = lane >> 3;
  float alpha = ex[(size_t)e * HEADS + h] /
                (denom[s * HEADS + h] + 1e-16f) * cnt[s];
  const float4 mv = *(const float4*)(Mn + (size_t)s * EMB + 4 * lane);
  const float4 me = *(const float4*)(emb + (size_t)c * EMB + 4 * lane);
  float* dp = aggr + (size_t)d * EMB + 4 * lane;
  atomicAdd(dp + 0, (mv.x + me.x) * alpha);
  atomicAdd(dp + 1, (mv.y + me.y) * alpha);
  atomicAdd(dp + 2, (mv.z + me.z) * alpha);
  atomicAdd(dp + 3, (mv.w + me.w) * alpha);
}

// ---------------------------------------------------------------------------
// 7: output BN stats (block-partial column sums), finalize, apply
// ---------------------------------------------------------------------------
__global__ void out_stats_kernel(const float* __restrict__ t1, float* __restrict__ osum,
                                 float* __restrict__ osumsq, int N) {
  int col = threadIdx.x;  // 128 threads
  float s = 0.0f, ss = 0.0f;
  for (int r = blockIdx.x; r < N; r += gridDim.x) {
    float v = t1[(size_t)r * EMB + col];
    s += v;
    ss += v * v;
  }
  atomicAdd(&osum[col], s);
  atomicAdd(&osumsq[col], ss);
}

__global__ void out_bn_finalize_kernel(const float* __restrict__ osum,
                                       const float* __restrict__ osumsq,
                                       float* __restrict__ omean, float* __restrict__ orstd,
                                       int N) {
  int col = threadIdx.x;
  float inv = 1.0f / (float)N;
  float m = osum[col] * inv;
  float var = osumsq[col] * inv - m * m;
  var = var > 0.0f ? var : 0.0f;
  omean[col] = m;
  orstd[col] = rsqrtf(var + 1e-5f);
}

__global__ void out_bn_apply_kernel(float* __restrict__ t1, const float* __restrict__ omean,
                                    const float* __restrict__ orstd,
                                    const float* __restrict__ go, const float* __restrict__ bbo,
                                    int N) {
  int idx = blockIdx.x * blockDim.x + threadIdx.x;
  if (idx >= N * EMB) return;
  int col = idx & 127;
  float v = (t1[idx] - omean[col]) * orstd[col] * go[col] + bbo[col];
  t1[idx] = v > 0.0f ? v : 0.0f;
}

// ---------------------------------------------------------------------------
// host
// ---------------------------------------------------------------------------
static void launch_gemm(const float* A, const float* A2, int lda, const float* B,
                        const float* bias, float scale, float* C, int M, int K,
                        hipStream_t stream) {
  int tiles = M / 16;
  int blocks = (tiles + 7) / 8;
  wmma_gemm_kernel<<<blocks, 256, 0, stream>>>(A, A2, lda, B, bias, scale, C, M, K);
}

extern "C" void kernel_launch(void* const* d_in, const int* in_sizes, int n_in,
                              void* d_out, int out_size, void* d_ws, size_t ws_size,
                              hipStream_t stream) {
  const float* x    = (const float*)d_in[0];
  const float* nfe  = (const float*)d_in[1];
  const int*   eidx = (const int*)d_in[2];
  const int*   ety  = (const int*)d_in[3];
  const int*   nty  = (const int*)d_in[4];
  const float* We1  = (const float*)d_in[5];
  const float* be1  = (const float*)d_in[6];
  const float* ge   = (const float*)d_in[7];
  const float* bbe  = (const float*)d_in[8];
  const float* We2  = (const float*)d_in[9];
  const float* be2  = (const float*)d_in[10];
  const float* Wk   = (const float*)d_in[11];
  const float* bk   = (const float*)d_in[12];
  const float* Wm   = (const float*)d_in[13];
  const float* bm   = (const float*)d_in[14];
  const float* Wq   = (const float*)d_in[15];
  const float* bq   = (const float*)d_in[16];
  const float* Wo1  = (const float*)d_in[17];
  const float* bo1  = (const float*)d_in[18];
  const float* go   = (const float*)d_in[19];
  const float* bbo  = (const float*)d_in[20];
  const float* Wo2  = (const float*)d_in[21];
  const float* bo2  = (const float*)d_in[22];

  const int N  = in_sizes[0] / EMB;
  const int E  = in_sizes[3];
  const int Et = E + N;
  const int* src0 = eidx;
  const int* dst0 = eidx + E;

  // workspace carve-up (floats)
  float* Wf = (float*)d_ws;
  size_t o = 0;
  float* Qn      = Wf + o; o += (size_t)N * EMB;
  float* Kn      = Wf + o; o += (size_t)N * EMB;
  float* Mn      = Wf + o; o += (size_t)N * EMB;
  float* aggr    = Wf + o; o += (size_t)N * EMB;
  float* t1      = Wf + o; o += (size_t)N * EMB;
  float* h1t     = Wf + o; o += (size_t)NCOMBO * EMB;
  float* bnt     = Wf + o; o += (size_t)NCOMBO * EMB;
  float* eat     = Wf + o; o += (size_t)NCOMBO * EMB;
  float* ekb     = Wf + o; o += (size_t)NCOMBO * EMB;
  float* embt    = Wf + o; o += (size_t)NCOMBO * EMB;
  float* scores  = Wf + o; o += (size_t)Et * HEADS;
  float* smax    = Wf + o; o += (size_t)N * HEADS;
  float* denom   = Wf + o; o += (size_t)N * HEADS;
  float* cnt     = Wf + o; o += (size_t)N;
  float* ccount  = Wf + o; o += NCOMBO;
  float* emean   = Wf + o; o += EMB;
  float* erstd   = Wf + o; o += EMB;
  float* osum    = Wf + o; o += EMB;
  float* osumsq  = Wf + o; o += EMB;
  float* omean   = Wf + o; o += EMB;
  float* orstd   = Wf + o; o += EMB;
  int*   combo   = (int*)(Wf + o);

  const float qscale = 1.0f / sqrtf(32.0f);

  // 0: init accumulators
  {
    int total = N * EMB;
    init_kernel<<<(total + 255) / 256, 256, 0, stream>>>(cnt, smax, denom, aggr,
                                                         ccount, osum, osumsq, N);
  }
  // 1: combos + counts
  combo_kernel<<<(Et + 255) / 256, 256, 0, stream>>>(ety, nty, src0, dst0, combo,
                                                     ccount, cnt, E, Et);
  // 2-3: edge-encoder first layer + exact BN over Et rows, on 624-combo table
  h1_table_kernel<<<(NCOMBO * EMB + 255) / 256, 256, 0, stream>>>(We1, be1, h1t);
  edge_bn_stats_kernel<<<1, EMB, 0, stream>>>(h1t, ccount, emean, erstd, Et);
  edge_bn_apply_kernel<<<(NCOMBO * EMB + 255) / 256, 256, 0, stream>>>(h1t, emean, erstd,
                                                                       ge, bbe, bnt);
  // table GEMMs (624x128 @ 128x128) — WMMA f32
  launch_gemm(bnt, nullptr, EMB, We2, be2, 1.0f, eat, NCOMBO, EMB, stream);       // edge_attr
  launch_gemm(eat, nullptr, EMB, Wk + 256 * EMB, bk, 1.0f, ekb, NCOMBO, EMB, stream);
  launch_gemm(eat, nullptr, EMB, Wm + 256 * EMB, bm, 1.0f, embt, NCOMBO, EMB, stream);
  // node GEMMs ([N,256] = [x | nfe] @ [256,128]) — WMMA f32
  launch_gemm(x, nfe, EMB, Wq, bq, qscale, Qn, N, 2 * EMB, stream);               // q (scaled)
  launch_gemm(x, nfe, EMB, Wk, nullptr, 1.0f, Kn, N, 2 * EMB, stream);            // k node part
  launch_gemm(x, nfe, EMB, Wm, nullptr, 1.0f, Mn, N, 2 * EMB, stream);            // m node part
  // 4-6: attention
  scores_kernel<<<(Et + 7) / 8, 256, 0, stream>>>(Qn, Kn, ekb, combo, src0, dst0,
                                                  scores, smax, E, Et);
  exp_kernel<<<(Et * HEADS + 255) / 256, 256, 0, stream>>>(scores, smax, denom, src0, E, Et);
  aggregate_kernel<<<(Et + 7) / 8, 256, 0, stream>>>(Mn, embt, combo, scores, denom,
                                                     cnt, src0, dst0, aggr, E, Et);
  // 7: output MLP with training-mode BN
  launch_gemm(aggr, nullptr, EMB, Wo1, bo1, 1.0f, t1, N, EMB, stream);
  out_stats_kernel<<<256, EMB, 0, stream>>>(t1, osum, osumsq, N);
  out_bn_finalize_kernel<<<1, EMB, 0, stream>>>(osum, osumsq, omean, orstd, N);
  out_bn_apply_kernel<<<(N * EMB + 255) / 256, 256, 0, stream>>>(t1, omean, orstd, go, bbo, N);
  launch_gemm(t1, nullptr, EMB, Wo2, bo2, 1.0f, (float*)d_out, N, EMB, stream);
}